// CombinedLoss__3298534884130
// MI455X (gfx1250) — compile-verified
//
#include <hip/hip_runtime.h>

typedef float v2f __attribute__((ext_vector_type(2)));
typedef float v8f __attribute__((ext_vector_type(8)));

#define BATCH 4
#define NPTS  8192
#define MPTS  8192
#define COL_BLOCKS 4
#define COLS_PER_BLOCK (MPTS / COL_BLOCKS)              // 2048
#define COL_TILES (COLS_PER_BLOCK / 16)                 // 128
#define WAVES_PER_BLOCK 8
#define ROWTILES 2
#define ROWS_PER_WAVE (16 * ROWTILES)                   // 32
#define ROWS_PER_BLOCK (WAVES_PER_BLOCK * ROWS_PER_WAVE) // 256
#define ROW_BLOCKS (NPTS / ROWS_PER_BLOCK)              // 32
#define INF_BITS 0x7F800000u

// Fill min buffers with +inf bit pattern (nonneg-float ordering == uint ordering).
__global__ void init_min_kernel(unsigned* __restrict__ p, int n) {
    int i = blockIdx.x * blockDim.x + threadIdx.x;
    if (i < n) p[i] = INF_BITS;
}

// Wave = 2 x-row-tiles (32 rows) vs a 2048-column y slice.
// d2 tile computed inside v_wmma_f32_16x16x4_f32 via
//   A=[x0,x1,x2,|x|^2], B=[-2y0,-2y1,-2y2,1], C=|y|^2 (per-column broadcast).
__global__ __launch_bounds__(256) void chamfer_kernel(
    const float* __restrict__ X, const float* __restrict__ Y,
    unsigned* __restrict__ minx, unsigned* __restrict__ miny) {
    __shared__ unsigned colmin[COLS_PER_BLOCK];

    const int tid  = threadIdx.x;
    const int lane = tid & 31;
    const int wave = tid >> 5;
    const int cb   = blockIdx.x % COL_BLOCKS;
    const int rb   = (blockIdx.x / COL_BLOCKS) % ROW_BLOCKS;
    const int b    = blockIdx.x / (COL_BLOCKS * ROW_BLOCKS);

    for (int i = tid; i < COLS_PER_BLOCK; i += blockDim.x) colmin[i] = INF_BITS;
    __syncthreads();

    const float* xb = X + (size_t)b * NPTS * 3;
    const float* yb = Y + (size_t)b * MPTS * 3;
    const int rowbase  = rb * ROWS_PER_BLOCK + wave * ROWS_PER_WAVE;
    const int colbase0 = cb * COLS_PER_BLOCK;

    const int hi  = lane >> 4;   // 0: K=0,1 / rows 0..7 ; 1: K=2,3 / rows 8..15
    const int l16 = lane & 15;

    // A fragments (one per 16-row tile), built once.
    v2f afrag[ROWTILES];
    float rmin[ROWTILES][8];
#pragma unroll
    for (int j = 0; j < ROWTILES; ++j) {
        const int r = rowbase + j * 16 + l16;
        const float px = xb[r * 3 + 0];
        const float py = xb[r * 3 + 1];
        const float pz = xb[r * 3 + 2];
        const float xn2 = px * px + py * py + pz * pz;
        afrag[j].x = hi ? pz  : px;   // K=0 (lanes 0-15) / K=2 (lanes 16-31)
        afrag[j].y = hi ? xn2 : py;   // K=1 / K=3
#pragma unroll
        for (int v = 0; v < 8; ++v) rmin[j][v] = 3.4e38f;
    }

    // Software-pipelined y loads: each lane owns column l16 of each tile.
    const float* yp = yb + (size_t)(colbase0 + l16) * 3;
    float q0 = yp[0], q1 = yp[1], q2 = yp[2];

    for (int t = 0; t < COL_TILES; ++t) {
        const float cq0 = q0, cq1 = q1, cq2 = q2;
        if (t + 1 < COL_TILES) {      // prefetch next tile's point
            yp += 48;                 // 16 points * 3 floats
            q0 = yp[0]; q1 = yp[1]; q2 = yp[2];
        }
        const float yn2 = cq0 * cq0 + cq1 * cq1 + cq2 * cq2;

        v2f bfrag;
        bfrag.x = hi ? (-2.0f * cq2) : (-2.0f * cq0);  // K=2 row / K=0 row
        bfrag.y = hi ? 1.0f          : (-2.0f * cq1);  // K=3 row / K=1 row

        v8f d[ROWTILES];
#pragma unroll
        for (int j = 0; j < ROWTILES; ++j) {
            v8f cf;
#pragma unroll
            for (int v = 0; v < 8; ++v) cf[v] = yn2;   // |y_col|^2 broadcast
            d[j] = __builtin_amdgcn_wmma_f32_16x16x4_f32(
                false, afrag[j], false, bfrag, (short)0, cf, false, false);
        }

        // column min over 32 rows: min of raw d then clamp (monotone)
        float cmin = d[0][0];
#pragma unroll
        for (int j = 0; j < ROWTILES; ++j)
#pragma unroll
            for (int v = 0; v < 8; ++v) cmin = fminf(cmin, d[j][v]);
        cmin = fmaxf(cmin, 0.0f);
        // all 32 lanes: 2 lanes per column address, DS atomic unit merges
        atomicMin(&colmin[t * 16 + l16], __float_as_uint(cmin));

        // running row mins: min(rmin, max(d,0)) == med3(0, d, rmin) since rmin>=0
#pragma unroll
        for (int j = 0; j < ROWTILES; ++j)
#pragma unroll
            for (int v = 0; v < 8; ++v)
                rmin[j][v] = __builtin_amdgcn_fmed3f(0.0f, d[j][v], rmin[j][v]);
    }

    // reduce row mins across the 16 lanes of each half (masks < 16 stay in-half)
#pragma unroll
    for (int mask = 1; mask <= 8; mask <<= 1)
#pragma unroll
        for (int j = 0; j < ROWTILES; ++j)
#pragma unroll
            for (int v = 0; v < 8; ++v)
                rmin[j][v] = fminf(rmin[j][v], __shfl_xor(rmin[j][v], mask, 32));

    if (l16 == 0) {
#pragma unroll
        for (int j = 0; j < ROWTILES; ++j)
#pragma unroll
            for (int v = 0; v < 8; ++v)
                atomicMin(&minx[(size_t)b * NPTS + rowbase + j * 16 + hi * 8 + v],
                          __float_as_uint(rmin[j][v]));
    }

    __syncthreads();
    for (int i = tid; i < COLS_PER_BLOCK; i += blockDim.x)
        atomicMin(&miny[(size_t)b * MPTS + colbase0 + i], colmin[i]);
}

__global__ __launch_bounds__(256) void finalize_kernel(
    const unsigned* __restrict__ minx, const unsigned* __restrict__ miny,
    const float* __restrict__ dens, float* __restrict__ out) {
    __shared__ float s0[256], s1[256], s2[256];
    const int tid = threadIdx.x;
    float sx = 0.f, sy = 0.f, sd = 0.f;
    for (int i = tid; i < BATCH * NPTS; i += 256) sx += __uint_as_float(minx[i]);
    for (int i = tid; i < BATCH * MPTS; i += 256) sy += __uint_as_float(miny[i]);
    for (int i = tid; i < BATCH * NPTS; i += 256) sd += fabsf(dens[i]);
    s0[tid] = sx; s1[tid] = sy; s2[tid] = sd;
    __syncthreads();
    for (int k = 128; k > 0; k >>= 1) {
        if (tid < k) {
            s0[tid] += s0[tid + k];
            s1[tid] += s1[tid + k];
            s2[tid] += s2[tid + k];
        }
        __syncthreads();
    }
    if (tid == 0) {
        float cham = s0[0] / (float)(BATCH * NPTS) + s1[0] / (float)(BATCH * MPTS);
        cham = fminf(fmaxf(cham, 0.0f), 1.0e6f);
        float dl = s2[0] / (float)(BATCH * NPTS);
        out[0] = cham + 0.1f * dl;  // total
        out[1] = cham;              // chamfer_loss
        out[2] = dl;                // density_loss
    }
}

extern "C" void kernel_launch(void* const* d_in, const int* in_sizes, int n_in,
                              void* d_out, int out_size, void* d_ws, size_t ws_size,
                              hipStream_t stream) {
    const float* X = (const float*)d_in[0];   // pred_points   [4,8192,3]
    const float* Y = (const float*)d_in[1];   // target_points [4,8192,3]
    const float* D = (const float*)d_in[2];   // pred_densities[4,8192]
    float* out = (float*)d_out;

    unsigned* minx = (unsigned*)d_ws;             // B*N floats (as bits)
    unsigned* miny = minx + BATCH * NPTS;         // B*M floats (as bits)

    const int initn = BATCH * NPTS + BATCH * MPTS;
    init_min_kernel<<<(initn + 255) / 256, 256, 0, stream>>>(minx, initn);

    chamfer_kernel<<<BATCH * ROW_BLOCKS * COL_BLOCKS, 256, 0, stream>>>(
        X, Y, minx, miny);

    finalize_kernel<<<1, 256, 0, stream>>>(minx, miny, D, out);
}